// SAGEConvNet_5566277616459
// MI455X (gfx1250) — compile-verified
//
#include <hip/hip_runtime.h>
#include <hip/hip_bf16.h>
#include <math.h>

typedef float v2f __attribute__((ext_vector_type(2)));
typedef float v8f __attribute__((ext_vector_type(8)));

#define HID 64

// ---------------------------------------------------------------------------
// Dual GEMM: T = X @ Wl^T            (no bias)
//            S = X @ Wr^T + bias
// X: [N, Fin] row-major; Wl/Wr: [64, Fin] row-major; T/S: [N, 64].
// One wave per 16-row tile; 4 WMMA accumulators cover the 64-wide output.
// Weights staged in LDS (one matrix at a time, <= 32 KB).
// Uses V_WMMA_F32_16X16X4_F32 (exact fp32 matrix math).
// ---------------------------------------------------------------------------
__global__ void __launch_bounds__(256)
sage_gemm_dual(const float* __restrict__ X,
               const float* __restrict__ Wl,
               const float* __restrict__ Wr,
               const float* __restrict__ bias,
               float* __restrict__ T,
               float* __restrict__ S,
               int N, int Fin)
{
    __shared__ float ldsW[64 * 128];   // 32 KB max (Fin<=128)

    const int tid   = threadIdx.x;
    const int wave  = tid >> 5;
    const int lane  = tid & 31;
    const int hsel  = lane >> 4;        // 0: K pair {0,1}; 1: K pair {2,3}
    const int l15   = lane & 15;
    const int tileM = blockIdx.x * 8 + wave;
    const int row   = tileM * 16 + l15; // A-matrix: M = lane&15 (both halves)

    for (int p = 0; p < 2; ++p) {
        const float* __restrict__ W = p ? Wr : Wl;
        // cooperative stage of one 64 x Fin weight matrix into LDS
        for (int i = tid; i < 64 * Fin; i += 256) ldsW[i] = W[i];
        __syncthreads();

        if (tileM * 16 < N) {           // wave-uniform: EXEC stays all-ones
            v8f acc0 = {}, acc1 = {}, acc2 = {}, acc3 = {};
            for (int kk = 0; kk < Fin; kk += 4) {
                const int kb = kk + hsel * 2;
                v2f a;
                a.x = X[(size_t)row * Fin + kb];
                a.y = X[(size_t)row * Fin + kb + 1];

                // B fragment: B[k][n] = W[n][k]; n = nt*16 + (lane&15)
                v2f b0, b1, b2, b3;
                b0.x = ldsW[(0 * 16 + l15) * Fin + kb];
                b0.y = ldsW[(0 * 16 + l15) * Fin + kb + 1];
                b1.x = ldsW[(1 * 16 + l15) * Fin + kb];
                b1.y = ldsW[(1 * 16 + l15) * Fin + kb + 1];
                b2.x = ldsW[(2 * 16 + l15) * Fin + kb];
                b2.y = ldsW[(2 * 16 + l15) * Fin + kb + 1];
                b3.x = ldsW[(3 * 16 + l15) * Fin + kb];
                b3.y = ldsW[(3 * 16 + l15) * Fin + kb + 1];

                acc0 = __builtin_amdgcn_wmma_f32_16x16x4_f32(false, a, false, b0,
                                                             (short)0, acc0, false, false);
                acc1 = __builtin_amdgcn_wmma_f32_16x16x4_f32(false, a, false, b1,
                                                             (short)0, acc1, false, false);
                acc2 = __builtin_amdgcn_wmma_f32_16x16x4_f32(false, a, false, b2,
                                                             (short)0, acc2, false, false);
                acc3 = __builtin_amdgcn_wmma_f32_16x16x4_f32(false, a, false, b3,
                                                             (short)0, acc3, false, false);
            }

            float* __restrict__ Out = p ? S : T;
            v8f accs[4] = {acc0, acc1, acc2, acc3};
            #pragma unroll
            for (int nt = 0; nt < 4; ++nt) {
                const int ncol = nt * 16 + l15;
                const float bv = p ? bias[ncol] : 0.0f;
                #pragma unroll
                for (int i = 0; i < 8; ++i) {
                    // C/D layout: VGPR i -> M = i (lanes 0-15) / i+8 (lanes 16-31)
                    const int m = i + hsel * 8;
                    Out[(size_t)(tileM * 16 + m) * HID + ncol] = accs[nt][i] + bv;
                }
            }
        }
        __syncthreads();
    }
}

// ---------------------------------------------------------------------------
// Output GEMM: O[N,40] = H[N,64] @ Wout[40,64]^T + bout.  Padded to 48 cols.
// ---------------------------------------------------------------------------
__global__ void __launch_bounds__(256)
sage_gemm_out(const float* __restrict__ H,
              const float* __restrict__ Wout,
              const float* __restrict__ bout,
              float* __restrict__ O, int N)
{
    __shared__ float ldsW[48 * 64];    // 12 KB, zero-padded rows 40..47

    const int tid = threadIdx.x;
    for (int i = tid; i < 48 * 64; i += 256) {
        const int r = i >> 6, c = i & 63;
        ldsW[i] = (r < 40) ? Wout[r * 64 + c] : 0.0f;
    }
    __syncthreads();

    const int wave  = tid >> 5;
    const int lane  = tid & 31;
    const int hsel  = lane >> 4;
    const int l15   = lane & 15;
    const int tileM = blockIdx.x * 8 + wave;
    if (tileM * 16 >= N) return;       // wave-uniform

    const int row = tileM * 16 + l15;
    v8f acc0 = {}, acc1 = {}, acc2 = {};
    for (int kk = 0; kk < 64; kk += 4) {
        const int kb = kk + hsel * 2;
        v2f a;
        a.x = H[(size_t)row * HID + kb];
        a.y = H[(size_t)row * HID + kb + 1];
        v2f b0, b1, b2;
        b0.x = ldsW[(0 * 16 + l15) * 64 + kb];
        b0.y = ldsW[(0 * 16 + l15) * 64 + kb + 1];
        b1.x = ldsW[(1 * 16 + l15) * 64 + kb];
        b1.y = ldsW[(1 * 16 + l15) * 64 + kb + 1];
        b2.x = ldsW[(2 * 16 + l15) * 64 + kb];
        b2.y = ldsW[(2 * 16 + l15) * 64 + kb + 1];
        acc0 = __builtin_amdgcn_wmma_f32_16x16x4_f32(false, a, false, b0,
                                                     (short)0, acc0, false, false);
        acc1 = __builtin_amdgcn_wmma_f32_16x16x4_f32(false, a, false, b1,
                                                     (short)0, acc1, false, false);
        acc2 = __builtin_amdgcn_wmma_f32_16x16x4_f32(false, a, false, b2,
                                                     (short)0, acc2, false, false);
    }

    v8f accs[3] = {acc0, acc1, acc2};
    #pragma unroll
    for (int nt = 0; nt < 3; ++nt) {
        const int ncol = nt * 16 + l15;
        if (ncol < 40) {
            const float bv = bout[ncol];
            #pragma unroll
            for (int i = 0; i < 8; ++i) {
                const int m = i + hsel * 8;
                O[(size_t)(tileM * 16 + m) * 40 + ncol] = accs[nt][i] + bv;
            }
        }
    }
}

// ---------------------------------------------------------------------------
// Degree: deg[d] += 1 for each edge (fp32 hardware atomics, computed once).
// ---------------------------------------------------------------------------
__global__ void __launch_bounds__(256)
sage_degree(const int* __restrict__ dst, float* __restrict__ deg, int E)
{
    const int e = blockIdx.x * 256 + threadIdx.x;
    if (e < E) unsafeAtomicAdd(&deg[dst[e]], 1.0f);
}

// ---------------------------------------------------------------------------
// Edge scatter: AGG[dst] += T[src] (64 floats). 16 lanes/edge, float4 each.
// T and AGG are L2-resident (25.6 MB each vs 192 MB L2).
// ---------------------------------------------------------------------------
__global__ void __launch_bounds__(256)
sage_scatter(const float* __restrict__ T,
             const int* __restrict__ src,
             const int* __restrict__ dst,
             float* __restrict__ AGG, int E)
{
    const long long tid = (long long)blockIdx.x * 256 + threadIdx.x;
    const long long e   = tid >> 4;
    const int       lw  = (int)(tid & 15);
    if (e >= E) return;
    const int s = src[e];
    const int d = dst[e];
    const float4 v = *(const float4*)(T + (size_t)s * HID + lw * 4);
    float* p = AGG + (size_t)d * HID + lw * 4;
    unsafeAtomicAdd(p + 0, v.x);
    unsafeAtomicAdd(p + 1, v.y);
    unsafeAtomicAdd(p + 2, v.z);
    unsafeAtomicAdd(p + 3, v.w);
}

// ---------------------------------------------------------------------------
// Epilogue: H = [elu](AGG / max(deg,1) + S)
// ---------------------------------------------------------------------------
__global__ void __launch_bounds__(256)
sage_finalize(const float* __restrict__ AGG,
              const float* __restrict__ S,
              const float* __restrict__ deg,
              float* __restrict__ H, int N, int do_elu)
{
    const int t = blockIdx.x * 256 + threadIdx.x;
    if (t >= N * HID) return;
    const int n = t >> 6;
    float dg = deg[n];
    dg = dg > 1.0f ? dg : 1.0f;
    float v = AGG[t] / dg + S[t];
    if (do_elu) v = (v > 0.0f) ? v : expm1f(v);
    H[t] = v;
}

extern "C" void kernel_launch(void* const* d_in, const int* in_sizes, int n_in,
                              void* d_out, int out_size, void* d_ws, size_t ws_size,
                              hipStream_t stream)
{
    const float* x    = (const float*)d_in[0];
    const int*   ei   = (const int*)d_in[1];
    const float* W1l  = (const float*)d_in[2];
    const float* b1   = (const float*)d_in[3];
    const float* W1r  = (const float*)d_in[4];
    const float* W2l  = (const float*)d_in[5];
    const float* b2   = (const float*)d_in[6];
    const float* W2r  = (const float*)d_in[7];
    const float* W3l  = (const float*)d_in[8];
    const float* b3   = (const float*)d_in[9];
    const float* W3r  = (const float*)d_in[10];
    const float* Wout = (const float*)d_in[11];
    const float* bout = (const float*)d_in[12];

    const int N = in_sizes[0] / 128;     // 100000
    const int E = in_sizes[1] / 2;       // 1600000
    const int* src = ei;
    const int* dst = ei + E;

    float* ws        = (float*)d_ws;
    const size_t nh  = (size_t)N * HID;
    float* T   = ws;
    float* S   = ws + nh;
    float* AGG = ws + 2 * nh;
    float* H   = ws + 3 * nh;
    float* deg = ws + 4 * nh;

    const int tiles      = (N + 15) / 16;
    const int gemmBlocks = (tiles + 7) / 8;
    const int scatBlocks = (int)(((long long)E * 16 + 255) / 256);
    const int finBlocks  = (N * HID + 255) / 256;
    const int degBlocks  = (E + 255) / 256;

    // degree (shared across all 3 layers)
    hipMemsetAsync(deg, 0, (size_t)N * sizeof(float), stream);
    sage_degree<<<degBlocks, 256, 0, stream>>>(dst, deg, E);

    // ---- layer 1: Fin = 128
    sage_gemm_dual<<<gemmBlocks, 256, 0, stream>>>(x, W1l, W1r, b1, T, S, N, 128);
    hipMemsetAsync(AGG, 0, nh * sizeof(float), stream);
    sage_scatter<<<scatBlocks, 256, 0, stream>>>(T, src, dst, AGG, E);
    sage_finalize<<<finBlocks, 256, 0, stream>>>(AGG, S, deg, H, N, 1);

    // ---- layer 2: Fin = 64
    sage_gemm_dual<<<gemmBlocks, 256, 0, stream>>>(H, W2l, W2r, b2, T, S, N, 64);
    hipMemsetAsync(AGG, 0, nh * sizeof(float), stream);
    sage_scatter<<<scatBlocks, 256, 0, stream>>>(T, src, dst, AGG, E);
    sage_finalize<<<finBlocks, 256, 0, stream>>>(AGG, S, deg, H, N, 1);

    // ---- layer 3: Fin = 64, no ELU
    sage_gemm_dual<<<gemmBlocks, 256, 0, stream>>>(H, W3l, W3r, b3, T, S, N, 64);
    hipMemsetAsync(AGG, 0, nh * sizeof(float), stream);
    sage_scatter<<<scatBlocks, 256, 0, stream>>>(T, src, dst, AGG, E);
    sage_finalize<<<finBlocks, 256, 0, stream>>>(AGG, S, deg, H, N, 0);

    // ---- output projection 64 -> 40
    sage_gemm_out<<<gemmBlocks, 256, 0, stream>>>(H, Wout, bout, (float*)d_out, N);
}